// flow_layer_21749714387071
// MI455X (gfx1250) — compile-verified
//
#include <hip/hip_runtime.h>

#define C_CH 16  // channels per node

// ---------------------------------------------------------------------------
// Zero the per-(node,channel) float3 accumulator (9.6 MB) with float4 stores.
// ---------------------------------------------------------------------------
__global__ __launch_bounds__(256) void zero_f4(float4* __restrict__ p, int n4) {
    int i = blockIdx.x * 256 + threadIdx.x;
    if (i < n4) p[i] = make_float4(0.f, 0.f, 0.f, 0.f);
}

// ---------------------------------------------------------------------------
// One thread per (edge, channel). lane = channel within a half-wave, so the
// 16 lanes of each half-wave read one contiguous 192B channel-block per graph
// endpoint (coalesced, L2-resident). Scatter via hardware f32 global atomics
// (no-return global_atomic_add_f32, STOREcnt-tracked, resolved in L2).
//
//   log_p(q) = (theta/sin(theta)) * (q - <p,q> p),  theta = acos(clamp(<p,q>))
//   sin(acos(x)) == sqrt(1 - x^2) exactly; clamp keeps 1-x^2 >= ~2e-7.
// We accumulate +log; the sign is folded into the node pass.
// ---------------------------------------------------------------------------
__global__ __launch_bounds__(256) void edge_scatter(
    const float* __restrict__ x,
    const int*   __restrict__ senders,
    const int*   __restrict__ receivers,
    float*       __restrict__ acc,
    int n_edges) {
    int tid = blockIdx.x * 256 + threadIdx.x;
    int e = tid >> 4;
    if (e >= n_edges) return;
    int c = tid & 15;

#if __has_builtin(__builtin_prefetch)
    // gfx1250: global_prefetch_b8 — pull the streamed index arrays ahead.
    // Speculative prefetch past the end is silently dropped (ISA 10.5).
    if (threadIdx.x == 0) {
        __builtin_prefetch(senders + e + 16384);
        __builtin_prefetch(receivers + e + 16384);
    }
#endif

    // Index streams are single-use: non-temporal loads (th:NT) keep them from
    // displacing the hot 9.6 MB node tensor + accumulator in cache.
#if __has_builtin(__builtin_nontemporal_load)
    int s = __builtin_nontemporal_load(senders + e);
    int r = __builtin_nontemporal_load(receivers + e);
#else
    int s = senders[e];
    int r = receivers[e];
#endif

    const float* q = x + (s * C_CH + c) * 3;  // sender point
    const float* p = x + (r * C_CH + c) * 3;  // receiver point (base of log)
    float qx = q[0], qy = q[1], qz = q[2];
    float px = p[0], py = p[1], pz = p[2];

    float inner = fmaf(px, qx, fmaf(py, qy, pz * qz));
    inner = fminf(fmaxf(inner, -1.0f + 1e-7f), 1.0f - 1e-7f);
    float theta = acosf(inner);
    // theta / sin(theta) = theta * rsqrt(1 - inner^2)
    float k = theta * rsqrtf(fmaxf(1.0f - inner * inner, 1e-12f));

    float lx = k * fmaf(-inner, px, qx);
    float ly = k * fmaf(-inner, py, qy);
    float lz = k * fmaf(-inner, pz, qz);

    float* a = acc + (r * C_CH + c) * 3;
    unsafeAtomicAdd(a + 0, lx);   // -> global_atomic_add_f32 (no return)
    unsafeAtomicAdd(a + 1, ly);
    unsafeAtomicAdd(a + 2, lz);
}

// ---------------------------------------------------------------------------
// One thread per (node, channel):
//   lap = -acc;  keep = (||lap|| >= delta);  v = -t * (keep ? lap : 0)
//     => w = keep * t * acc
//   out = cos(||w||)*x + (sin(||w||)/||w||)*w   (||w|| floored at 1e-8)
// ---------------------------------------------------------------------------
__global__ __launch_bounds__(256) void node_update(
    const float* __restrict__ x,
    const float* __restrict__ acc,
    const float* __restrict__ t_sqrt,
    const float* __restrict__ delta_sqrt,
    float*       __restrict__ out,
    int n_nodes) {
    int tid = blockIdx.x * 256 + threadIdx.x;
    if (tid >= n_nodes * C_CH) return;
    int c = tid & 15;

    float ts = t_sqrt[c];
    float t  = ts * ts;            // N_STEPS == 1
    float dl = delta_sqrt[c];
    float delta = dl * dl;

    const float* a = acc + tid * 3;
    float ax = a[0], ay = a[1], az = a[2];
    const float* p = x + tid * 3;
    float px = p[0], py = p[1], pz = p[2];

    float n2   = fmaf(ax, ax, fmaf(ay, ay, az * az));
    float keep = (sqrtf(n2) >= delta) ? 1.0f : 0.0f;

    float wx = keep * t * ax;
    float wy = keep * t * ay;
    float wz = keep * t * az;

    float sq = fmaf(wx, wx, fmaf(wy, wy, wz * wz));
    float nr = sqrtf(fmaxf(sq, 1e-16f));
    float cn = cosf(nr);
    float sn = sinf(nr) / nr;

    float ox = fmaf(cn, px, sn * wx);
    float oy = fmaf(cn, py, sn * wy);
    float oz = fmaf(cn, pz, sn * wz);

    // Output is written once and never re-read on device: non-temporal stores.
#if __has_builtin(__builtin_nontemporal_store)
    __builtin_nontemporal_store(ox, out + tid * 3 + 0);
    __builtin_nontemporal_store(oy, out + tid * 3 + 1);
    __builtin_nontemporal_store(oz, out + tid * 3 + 2);
#else
    out[tid * 3 + 0] = ox;
    out[tid * 3 + 1] = oy;
    out[tid * 3 + 2] = oz;
#endif
}

// ---------------------------------------------------------------------------
// d_in order (setup_inputs): nodes f32[N*16*3], t_sqrt f32[16],
//                            delta_sqrt f32[16], senders i32[E], receivers i32[E]
// d_out: f32[N*16*3].  d_ws: accumulator (N*16*3 floats = 9.6 MB).
// ---------------------------------------------------------------------------
extern "C" void kernel_launch(void* const* d_in, const int* in_sizes, int n_in,
                              void* d_out, int out_size, void* d_ws, size_t ws_size,
                              hipStream_t stream) {
    const float* nodes      = (const float*)d_in[0];
    const float* t_sqrt     = (const float*)d_in[1];
    const float* delta_sqrt = (const float*)d_in[2];
    const int*   senders    = (const int*)d_in[3];
    const int*   receivers  = (const int*)d_in[4];
    float* out = (float*)d_out;
    float* acc = (float*)d_ws;

    int n_node_ch3 = in_sizes[0];              // n_nodes * 16 * 3 (divisible by 4)
    int n_nodes    = n_node_ch3 / (C_CH * 3);
    int n_edges    = in_sizes[3];

    int n4 = n_node_ch3 / 4;
    zero_f4<<<(n4 + 255) / 256, 256, 0, stream>>>((float4*)acc, n4);

    long total_ec = (long)n_edges * C_CH;
    int  eg = (int)((total_ec + 255) / 256);
    edge_scatter<<<eg, 256, 0, stream>>>(nodes, senders, receivers, acc, n_edges);

    int ng = (n_nodes * C_CH + 255) / 256;
    node_update<<<ng, 256, 0, stream>>>(nodes, acc, t_sqrt, delta_sqrt, out, n_nodes);
}